// GlobalLinearAttention_53523882442968
// MI455X (gfx1250) — compile-verified
//
#include <hip/hip_runtime.h>
#include <hip/hip_bf16.h>
#include <float.h>

// Problem constants (from reference)
#define B_   4
#define N_   8192
#define D_   512
#define H_   8
#define DH_  64
#define QKV_COLS (3 * H_ * DH_)   // 1536

typedef __attribute__((ext_vector_type(2))) float v2f;
typedef __attribute__((ext_vector_type(8))) float v8f;

// D = A(16x4 f32) * B(4x16 f32) + C(16x16 f32), wave32 WMMA (verified lowering)
static __device__ __forceinline__ v8f wmma4(v2f a, v2f b, v8f c) {
  return __builtin_amdgcn_wmma_f32_16x16x4_f32(
      /*neg_a=*/false, a, /*neg_b=*/false, b,
      /*c_mod=*/(short)0, c, /*reuse_a=*/false, /*reuse_b=*/false);
}

// Async global->LDS copy of 16 bytes per lane (gfx1250 ASYNCcnt path).
static __device__ __forceinline__ void async_cp16(unsigned lds_off,
                                                  const float* gaddr) {
  asm volatile("global_load_async_to_lds_b128 %0, %1, off"
               :: "v"(lds_off), "v"(gaddr)
               : "memory");
}
static __device__ __forceinline__ void wait_async0() {
  asm volatile("s_wait_asynccnt 0x0" ::: "memory");
}
static __device__ __forceinline__ unsigned lds_off(const void* p) {
  return (unsigned)(size_t)p;   // LDS generic addr: low 32 bits = LDS offset
}

// GEMM tiling
#define BM 128
#define BN 64
#define KC 32
#define ASTR 36   // padded LDS row stride for A (conflict-free, 16B aligned)
#define BSTR 80   // padded LDS row stride for B/k/v panels: halves of a wave
                  // land in disjoint bank sets (80 mod 64 = 16 -> rows 2 apart
                  // differ by 32 banks), 16B aligned (320 B rows)

// ---------------------------------------------------------------------------
// Kernel A: qkv = x @ W_qkv, LDS-staged (async, double buffered), 128x64 block
// tile, 8 waves x (32x32 register tile). Epilogue scatters into head-major
// q/k/v and applies the mask (k -> -FLT_MAX, v -> 0 on masked rows).
// M = B*N = 32768, K = 512, N = 1536.
// ---------------------------------------------------------------------------
__global__ void __launch_bounds__(256) qkv_gemm(
    const float* __restrict__ x, const float* __restrict__ Wqkv,
    const unsigned char* __restrict__ mask, float* __restrict__ q,
    float* __restrict__ k, float* __restrict__ v) {
  __shared__ float aLds[2][BM * ASTR];  // 36864 B
  __shared__ float bLds[2][KC * BSTR];  // 20480 B

  const int tid  = threadIdx.x;
  const int lane = tid & 31;
  const int wave = tid >> 5;
  const int half = lane >> 4;
  const int l15  = lane & 15;
  const int m0g  = blockIdx.x * BM;
  const int n0g  = blockIdx.y * BN;
  const int wm   = (wave >> 1) * 32;
  const int wn   = (wave & 1) * 32;

  auto stage = [&](int buf, int kk0) {
    const int ar = tid >> 3;
    const int ac = (tid & 7) * 4;
#pragma unroll
    for (int i = 0; i < 4; ++i) {
      const int row = ar + i * 32;
      async_cp16(lds_off(&aLds[buf][row * ASTR + ac]),
                 x + (size_t)(m0g + row) * D_ + kk0 + ac);
    }
    const int br = tid >> 4;
    const int bc = (tid & 15) * 4;
#pragma unroll
    for (int i = 0; i < 2; ++i) {
      const int row = br + i * 16;
      async_cp16(lds_off(&bLds[buf][row * BSTR + bc]),
                 Wqkv + (size_t)(kk0 + row) * QKV_COLS + n0g + bc);
    }
  };

  stage(0, 0);
  v8f acc[2][2] = {};
  int buf = 0;
  for (int kk0 = 0; kk0 < D_; kk0 += KC) {
    wait_async0();
    __syncthreads();
    if (kk0 + KC < D_) stage(buf ^ 1, kk0 + KC);
    const float* A  = aLds[buf];
    const float* Bt = bLds[buf];
#pragma unroll
    for (int kk = 0; kk < KC; kk += 4) {
      const int kc = kk + half * 2;
      v2f a0, a1, b0, b1;
      a0.x = A[(wm + l15) * ASTR + kc];
      a0.y = A[(wm + l15) * ASTR + kc + 1];
      a1.x = A[(wm + 16 + l15) * ASTR + kc];
      a1.y = A[(wm + 16 + l15) * ASTR + kc + 1];
      b0.x = Bt[kc * BSTR + wn + l15];
      b0.y = Bt[(kc + 1) * BSTR + wn + l15];
      b1.x = Bt[kc * BSTR + wn + 16 + l15];
      b1.y = Bt[(kc + 1) * BSTR + wn + 16 + l15];
      acc[0][0] = wmma4(a0, b0, acc[0][0]);
      acc[0][1] = wmma4(a0, b1, acc[0][1]);
      acc[1][0] = wmma4(a1, b0, acc[1][0]);
      acc[1][1] = wmma4(a1, b1, acc[1][1]);
    }
    buf ^= 1;
  }

#pragma unroll
  for (int sm = 0; sm < 2; ++sm) {
#pragma unroll
    for (int sn = 0; sn < 2; ++sn) {
      const int col  = n0g + wn + sn * 16 + l15;  // 0..1535
      const int part = col / 512;                 // 0=q 1=k 2=v
      const int f    = col & 511;
      const int h    = f >> 6;
      const int d    = f & 63;
#pragma unroll
      for (int i = 0; i < 8; ++i) {
        const int row = m0g + wm + sm * 16 + i + half * 8;
        const int bb  = row >> 13;
        const int n   = row & (N_ - 1);
        const size_t idx = ((size_t)(bb * H_ + h) * N_ + n) * DH_ + d;
        const float val = acc[sm][sn][i];
        if (part == 0) {
          q[idx] = val;
        } else if (part == 1) {
          k[idx] = mask[bb * N_ + n] ? val : -FLT_MAX;
        } else {
          v[idx] = mask[bb * N_ + n] ? val : 0.0f;
        }
      }
    }
  }
}

// ---------------------------------------------------------------------------
// Kernel B: q <- softmax(q, axis=-1) * DH^-0.5 ; one wave32 per 64-elem row.
// ---------------------------------------------------------------------------
__global__ void q_softmax(float* __restrict__ q) {
  const int lane = threadIdx.x & 31;
  const int wave = threadIdx.x >> 5;
  const long long row = (long long)blockIdx.x * 8 + wave;
  float* p = q + row * DH_;
  float a0 = p[lane];
  float a1 = p[lane + 32];
  float mx = fmaxf(a0, a1);
#pragma unroll
  for (int off = 16; off > 0; off >>= 1) mx = fmaxf(mx, __shfl_xor(mx, off, 32));
  const float e0 = __expf(a0 - mx);
  const float e1 = __expf(a1 - mx);
  float s = e0 + e1;
#pragma unroll
  for (int off = 16; off > 0; off >>= 1) s += __shfl_xor(s, off, 32);
  const float inv = 0.125f / s;   // scale = DH^-0.5 = 1/8
  p[lane]      = e0 * inv;
  p[lane + 32] = e1 * inv;
}

// ---------------------------------------------------------------------------
// Kernel C: k <- softmax(k, axis=-2) per (b,h) column d over N=8192.
// Online (single-pass) max+sum, then one normalize pass. Masked entries are
// -FLT_MAX -> exp() == 0 (all-masked column degenerates to uniform, matching
// the reference).
// ---------------------------------------------------------------------------
__global__ void k_colsoftmax(float* __restrict__ k) {
  __shared__ float redM[256];
  __shared__ float redS[256];
  __shared__ float Md[64];
  __shared__ float Sd[64];
  const int tid = threadIdx.x;
  const int d = tid & 63;
  const int g = tid >> 6;  // 0..3
  float* base = k + (size_t)blockIdx.x * N_ * DH_;

  float mx = -FLT_MAX;
  float s  = 0.0f;
  for (int n = g; n < N_; n += 4) {
    const float val = base[(size_t)n * DH_ + d];
    const float mn  = fmaxf(mx, val);
    s = s * __expf(mx - mn) + __expf(val - mn);
    mx = mn;
  }
  redM[tid] = mx;
  redS[tid] = s;
  __syncthreads();
  if (g == 0) {
    float m01 = fmaxf(redM[d], redM[64 + d]);
    float m23 = fmaxf(redM[128 + d], redM[192 + d]);
    float M   = fmaxf(m01, m23);
    float S = redS[d] * __expf(redM[d] - M) +
              redS[64 + d] * __expf(redM[64 + d] - M) +
              redS[128 + d] * __expf(redM[128 + d] - M) +
              redS[192 + d] * __expf(redM[192 + d] - M);
    Md[d] = M;
    Sd[d] = S;
  }
  __syncthreads();
  const float m = Md[d];
  const float inv = 1.0f / Sd[d];
  for (int n = g; n < N_; n += 4) {
    const size_t idx = (size_t)n * DH_ + d;
    base[idx] = __expf(base[idx] - m) * inv;
  }
}

// ---------------------------------------------------------------------------
// Zero a float buffer (workspace is poisoned with 0xAA by the harness).
// ---------------------------------------------------------------------------
__global__ void zero_buf(float* __restrict__ p, int count) {
  const int i = blockIdx.x * 256 + threadIdx.x;
  if (i < count) p[i] = 0.0f;
}

// ---------------------------------------------------------------------------
// Kernel D: context[b][h] = k_sm^T (64xN) @ v (Nx64). K split over 16 grid
// chunks; k/v panels async-staged to LDS (double buffered) so each block
// reads them exactly once; partial tiles accumulated with
// global_atomic_add_f32. Wave computes subtiles (md,ne) and (md,ne+16),
// sharing the A fragment.
// ---------------------------------------------------------------------------
#define CNK 32
__global__ void __launch_bounds__(256) context_gemm(
    const float* __restrict__ k, const float* __restrict__ v,
    float* __restrict__ ctx) {
  __shared__ float kLds[2][CNK * BSTR];  // 20480 B
  __shared__ float vLds[2][CNK * BSTR];  // 20480 B
  const int tid  = threadIdx.x;
  const int lane = tid & 31;
  const int wave = tid >> 5;
  const int half = lane >> 4;
  const int l15  = lane & 15;
  const int bh = blockIdx.x;     // 0..31
  const int chunk = blockIdx.y;  // 0..15
  const float* kb = k + (size_t)bh * N_ * DH_;
  const float* vb = v + (size_t)bh * N_ * DH_;
  const int n_beg = chunk * (N_ / 16);

  const int md = (wave >> 1) * 16;   // A-row block (d)
  const int ne = (wave & 1) * 32;    // B-col block (e), two subtiles ne, ne+16

  auto stage = [&](int buf, int nb) {
    const int r = tid >> 4;          // 0..15
    const int c = (tid & 15) * 4;    // 0..60
#pragma unroll
    for (int i = 0; i < 2; ++i) {
      const int row = r + i * 16;
      async_cp16(lds_off(&kLds[buf][row * BSTR + c]),
                 kb + (size_t)(nb + row) * DH_ + c);
      async_cp16(lds_off(&vLds[buf][row * BSTR + c]),
                 vb + (size_t)(nb + row) * DH_ + c);
    }
  };

  stage(0, n_beg);
  v8f acc[2] = {};
  int buf = 0;
  for (int nb = 0; nb < N_ / 16; nb += CNK) {
    wait_async0();
    __syncthreads();
    if (nb + CNK < N_ / 16) stage(buf ^ 1, n_beg + nb + CNK);
    const float* K = kLds[buf];
    const float* V = vLds[buf];
#pragma unroll
    for (int nn = 0; nn < CNK; nn += 4) {
      const int n2 = nn + half * 2;
      v2f a, b0, b1;
      a.x  = K[n2 * BSTR + md + l15];        // A[d][n] = k_sm[n][d]
      a.y  = K[(n2 + 1) * BSTR + md + l15];
      b0.x = V[n2 * BSTR + ne + l15];
      b0.y = V[(n2 + 1) * BSTR + ne + l15];
      b1.x = V[n2 * BSTR + ne + 16 + l15];
      b1.y = V[(n2 + 1) * BSTR + ne + 16 + l15];
      acc[0] = wmma4(a, b0, acc[0]);
      acc[1] = wmma4(a, b1, acc[1]);
    }
    buf ^= 1;
  }

  float* cb = ctx + (size_t)bh * DH_ * DH_;
#pragma unroll
  for (int i = 0; i < 8; ++i) {
    atomicAdd(&cb[(md + i + half * 8) * DH_ + ne + l15], acc[0][i]);
    atomicAdd(&cb[(md + i + half * 8) * DH_ + ne + 16 + l15], acc[1][i]);
  }
}

// ---------------------------------------------------------------------------
// Kernel E: CW[b][h] (64x512) = context[b][h] (64x64) @ W_out[h*64:(h+1)*64,:]
// Stacked over h this gives a contiguous per-batch 512x512 matrix.
// ---------------------------------------------------------------------------
__global__ void cw_gemm(const float* __restrict__ ctx,
                        const float* __restrict__ Wout,
                        float* __restrict__ cw) {
  const int lane = threadIdx.x & 31;
  const int wave = threadIdx.x >> 5;
  const int bh = blockIdx.x;
  const int h  = bh & 7;
  const int n0 = blockIdx.y * 128 + wave * 16;
  const int half = lane >> 4;
  const int l15  = lane & 15;
  const float* cb = ctx + (size_t)bh * DH_ * DH_;
  const float* wb = Wout + (size_t)h * DH_ * D_;
  float* ob = cw + (size_t)bh * DH_ * D_;

  for (int mt = 0; mt < 4; ++mt) {
    const int m0 = mt * 16;
    v8f acc = {};
    for (int kk = 0; kk < DH_; kk += 4) {
      v2f a, b;
      const float* pa = cb + (m0 + l15) * DH_ + kk + half * 2;
      a.x = pa[0];
      a.y = pa[1];
      const float* pb = wb + (size_t)(kk + half * 2) * D_ + n0 + l15;
      b.x = pb[0];
      b.y = pb[D_];
      acc = wmma4(a, b, acc);
    }
#pragma unroll
    for (int i = 0; i < 8; ++i)
      ob[(m0 + i + half * 8) * D_ + n0 + l15] = acc[i];
  }
}

// ---------------------------------------------------------------------------
// Kernel F: out[b] = Q_cat[b] (N x 512) @ CW_stack[b] (512x512) + b_out,
// fusing attn = q@context with out = attn@W_out. Same LDS-staged scheme as
// kernel A; the A panel is gathered from head-major q by index arithmetic.
// ---------------------------------------------------------------------------
__global__ void __launch_bounds__(256) out_gemm(
    const float* __restrict__ q, const float* __restrict__ cw,
    const float* __restrict__ bout, float* __restrict__ out) {
  __shared__ float aLds[2][BM * ASTR];
  __shared__ float bLds[2][KC * BSTR];

  const int tid  = threadIdx.x;
  const int lane = tid & 31;
  const int wave = tid >> 5;
  const int half = lane >> 4;
  const int l15  = lane & 15;
  const int b    = blockIdx.x >> 6;
  const int m0g  = (blockIdx.x & 63) * BM;
  const int n0g  = blockIdx.y * BN;
  const int wm   = (wave >> 1) * 32;
  const int wn   = (wave & 1) * 32;
  const float* cwb = cw + (size_t)b * D_ * D_;

  auto stage = [&](int buf, int kk0) {
    const int ar = tid >> 3;
    const int ac = (tid & 7) * 4;
#pragma unroll
    for (int i = 0; i < 4; ++i) {
      const int row = ar + i * 32;
      const int ck  = kk0 + ac;
      const int h0  = ck >> 6;
      const int d0  = ck & 63;
      async_cp16(lds_off(&aLds[buf][row * ASTR + ac]),
                 q + ((size_t)(b * H_ + h0) * N_ + (m0g + row)) * DH_ + d0);
    }
    const int br = tid >> 4;
    const int bc = (tid & 15) * 4;
#pragma unroll
    for (int i = 0; i < 2; ++i) {
      const int row = br + i * 16;
      async_cp16(lds_off(&bLds[buf][row * BSTR + bc]),
                 cwb + (size_t)(kk0 + row) * D_ + n0g + bc);
    }
  };

  stage(0, 0);
  v8f acc[2][2] = {};
  int buf = 0;
  for (int kk0 = 0; kk0 < D_; kk0 += KC) {
    wait_async0();
    __syncthreads();
    if (kk0 + KC < D_) stage(buf ^ 1, kk0 + KC);
    const float* A  = aLds[buf];
    const float* Bt = bLds[buf];
#pragma unroll
    for (int kk = 0; kk < KC; kk += 4) {
      const int kc = kk + half * 2;
      v2f a0, a1, b0, b1;
      a0.x = A[(wm + l15) * ASTR + kc];
      a0.y = A[(wm + l15) * ASTR + kc + 1];
      a1.x = A[(wm + 16 + l15) * ASTR + kc];
      a1.y = A[(wm + 16 + l15) * ASTR + kc + 1];
      b0.x = Bt[kc * BSTR + wn + l15];
      b0.y = Bt[(kc + 1) * BSTR + wn + l15];
      b1.x = Bt[kc * BSTR + wn + 16 + l15];
      b1.y = Bt[(kc + 1) * BSTR + wn + 16 + l15];
      acc[0][0] = wmma4(a0, b0, acc[0][0]);
      acc[0][1] = wmma4(a0, b1, acc[0][1]);
      acc[1][0] = wmma4(a1, b0, acc[1][0]);
      acc[1][1] = wmma4(a1, b1, acc[1][1]);
    }
    buf ^= 1;
  }

#pragma unroll
  for (int sm = 0; sm < 2; ++sm) {
#pragma unroll
    for (int sn = 0; sn < 2; ++sn) {
      const int j = n0g + wn + sn * 16 + l15;
      const float bj = bout[j];
#pragma unroll
      for (int i = 0; i < 8; ++i) {
        const int n = m0g + wm + sm * 16 + i + half * 8;
        out[((size_t)b * N_ + n) * D_ + j] = acc[sm][sn][i] + bj;
      }
    }
  }
}

// ---------------------------------------------------------------------------
extern "C" void kernel_launch(void* const* d_in, const int* in_sizes, int n_in,
                              void* d_out, int out_size, void* d_ws,
                              size_t ws_size, hipStream_t stream) {
  const float* x    = (const float*)d_in[0];
  const float* Wqkv = (const float*)d_in[1];
  const float* Wout = (const float*)d_in[2];
  const float* bout = (const float*)d_in[3];
  const unsigned char* mask = (const unsigned char*)d_in[4];  // jax bool = 1B
  float* out = (float*)d_out;

  // Workspace layout (floats): q | k | v | ctx | cw  (~196.5 MB total)
  const size_t HEADSZ = (size_t)B_ * H_ * N_ * DH_;  // 16M floats each
  float* q   = (float*)d_ws;
  float* k   = q + HEADSZ;
  float* v   = k + HEADSZ;
  float* ctx = v + HEADSZ;
  float* cw  = ctx + (size_t)B_ * H_ * DH_ * DH_;

  const dim3 blk(256);
  qkv_gemm<<<dim3(32768 / BM, QKV_COLS / BN), blk, 0, stream>>>(x, Wqkv, mask,
                                                                q, k, v);
  q_softmax<<<dim3((B_ * H_ * N_) / 8), blk, 0, stream>>>(q);
  k_colsoftmax<<<dim3(B_ * H_), blk, 0, stream>>>(k);
  zero_buf<<<dim3((B_ * H_ * DH_ * DH_ + 255) / 256), blk, 0, stream>>>(
      ctx, B_ * H_ * DH_ * DH_);
  context_gemm<<<dim3(B_ * H_, 16), blk, 0, stream>>>(k, v, ctx);
  cw_gemm<<<dim3(B_ * H_, 4), blk, 0, stream>>>(ctx, Wout, cw);
  out_gemm<<<dim3(B_ * (N_ / BM), D_ / BN), blk, 0, stream>>>(q, cw, bout, out);
}